// MultiGraphConvLayer_54099408060448
// MI455X (gfx1250) — compile-verified
//
#include <hip/hip_runtime.h>

typedef float v2f __attribute__((ext_vector_type(2)));
typedef float v8f __attribute__((ext_vector_type(8)));

#define N_NODES   100000
#define N_EDGES   800000
#define NF        81      // node features
#define NB        22      // bond features
#define NC        64      // out channels
#define KE        103     // NF + NB
#define KS_PAD    84      // NF padded to mult of 4  -> 21 K-steps
#define KE_PAD    104     // KE padded to mult of 4  -> 26 K-steps
#define RS_E      105     // edge LDS row stride (gcd(105,64)=1 -> conflict free)
#define RS_S      85      // self LDS row stride (gcd(85,64)=1 -> conflict free)

static __device__ __forceinline__ v8f wmma_f32(v2f a, v2f b, v8f c) {
    // D = A(16x4, f32) x B(4x16, f32) + C(16x16, f32)
    return __builtin_amdgcn_wmma_f32_16x16x4_f32(
        /*neg_a=*/false, a, /*neg_b=*/false, b,
        /*c_mod=*/(short)0, c, /*reuse_a=*/false, /*reuse_b=*/false);
}

static __device__ __forceinline__ void atomic_add_f32(float* p, float v) {
    unsafeAtomicAdd(p, v);   // global_atomic_add_f32, no-return
}

// ---------------------------------------------------------------------------
// Kernel 0: repack weights into WMMA B-fragment order in workspace.
// Row pairs are interleaved so one lane's B fragment {w[2p][col], w[2p+1][col]}
// is a single aligned 8-byte load; pad rows are physically zero.
//   ws2: [KS_PAD/2][NC] v2f    ws2[p][col] = { w_s[2p][col], w_s[2p+1][col] }
//   wn2: [KE_PAD/2][NC] v2f    wn2[p][col] = { w_n[2p][col], w_n[2p+1][col] }
// ---------------------------------------------------------------------------
__global__ __launch_bounds__(256) void pack_weights(
    const float* __restrict__ w_s, const float* __restrict__ w_n,
    v2f* __restrict__ ws2, v2f* __restrict__ wn2)
{
    const int i = blockIdx.x * 256 + threadIdx.x;
    if (i < (KS_PAD / 2) * NC) {
        const int p = i / NC, col = i - p * NC;
        const int r0 = 2 * p, r1 = r0 + 1;
        v2f v;
        v.x = (r0 < NF) ? w_s[r0 * NC + col] : 0.0f;
        v.y = (r1 < NF) ? w_s[r1 * NC + col] : 0.0f;
        ws2[i] = v;
    }
    if (i < (KE_PAD / 2) * NC) {
        const int p = i / NC, col = i - p * NC;
        const int r0 = 2 * p, r1 = r0 + 1;
        v2f v;
        v.x = (r0 < KE) ? w_n[r0 * NC + col] : 0.0f;
        v.y = (r1 < KE) ? w_n[r1 * NC + col] : 0.0f;
        wn2[i] = v;
    }
}

// ---------------------------------------------------------------------------
// Kernel 1: self_out = features @ w_s   (initializes d_out)
// 128 threads = 4 waves; each wave computes a 16(node) x 64(C) tile.
// ---------------------------------------------------------------------------
__global__ __launch_bounds__(128) void self_gemm(
    const float* __restrict__ feat,   // [N_NODES, NF]
    const v2f*   __restrict__ ws2,    // packed B fragments
    float* __restrict__ out)          // [N_NODES, NC]
{
    __shared__ float A[64 * RS_S];

    const int tid = threadIdx.x;
    const int nodeBase = blockIdx.x * 64;

    // zero K-pad columns 81..84
    for (int i = tid; i < 64 * 4; i += 128) {
        int r = i >> 2, c = 81 + (i & 3);
        A[r * RS_S + c] = 0.0f;
    }
    // stage 64 feature rows into LDS (guard tail block)
    for (int i = tid; i < 64 * NF; i += 128) {
        int r = i / NF, c = i - r * NF;
        int node = nodeBase + r;
        A[r * RS_S + c] = (node < N_NODES) ? feat[node * NF + c] : 0.0f;
    }
    __syncthreads();

    const int wave = tid >> 5;
    const int lane = tid & 31;
    const int half = lane >> 4;      // 0: lanes 0-15, 1: lanes 16-31
    const int m    = lane & 15;
    const int rowB = wave * 16;
    const v2f* bbase = ws2 + half * NC + m;   // + kk*2*NC + n*16 per fragment

    v8f acc[4] = {};
    #pragma unroll                    // straight-line: no acc register rotation
    for (int kk = 0; kk < KS_PAD / 4; ++kk) {
        const int kr0 = kk * 4 + 2 * half;     // A VGPR0 column
        v2f a;
        a.x = A[(rowB + m) * RS_S + kr0];
        a.y = A[(rowB + m) * RS_S + kr0 + 1];
        #pragma unroll
        for (int n = 0; n < 4; ++n) {
            const v2f b = bbase[kk * 2 * NC + n * 16];   // one global_load_b64
            acc[n] = wmma_f32(a, b, acc[n]);
        }
    }

    // D layout: vgpr v, lanes0-15 -> M=v ; lanes16-31 -> M=v+8 ; N = m (+tile)
    #pragma unroll
    for (int n = 0; n < 4; ++n) {
        #pragma unroll
        for (int v = 0; v < 8; ++v) {
            int node = nodeBase + rowB + v + 8 * half;
            if (node < N_NODES) out[node * NC + n * 16 + m] = acc[n][v];
        }
    }
}

// ---------------------------------------------------------------------------
// Kernel 2: per-edge fused feature build + edge_feat @ w_n + atomic scatter.
// 128 threads = 4 waves; block handles 128 edges; wave -> two 16-edge M tiles.
// ---------------------------------------------------------------------------
__global__ __launch_bounds__(128) void edge_gemm(
    const float* __restrict__ feat,   // [N_NODES, NF]
    const float* __restrict__ bond,   // [N_EDGES, NB]
    const v2f*   __restrict__ wn2,    // packed B fragments
    const int*   __restrict__ esrc,   // [N_EDGES]
    const int*   __restrict__ edst,   // [N_EDGES]
    float* __restrict__ out)          // [N_NODES, NC] (accumulated)
{
    __shared__ float E[128 * RS_E];
    __shared__ int   Src[128];

    const int tid = threadIdx.x;
    const int e   = blockIdx.x * 128 + tid;    // N_EDGES % 128 == 0

    // ---- Phase 1: one lane per edge builds its 103-wide row in LDS ----
    const int src = esrc[e];
    const int dst = edst[e];
    Src[tid] = src;
    const float* sf = feat + src * NF;
    const float* nf = feat + dst * NF;
    float* row = &E[tid * RS_E];

    const float n0 = nf[0], n1 = nf[1], n2 = nf[2];
    const float d0 = sf[0] - n0, d1 = sf[1] - n1, d2 = sf[2] - n2;
    const float inv = 1.0f / (d0 * d0 + d1 * d1 + d2 * d2);
    row[0] = n0 * inv; row[1] = n1 * inv; row[2] = n2 * inv;
    for (int f = 3; f < NF; ++f) row[f] = (sf[f] + nf[f]) * inv;
    const float* bd = bond + (long)e * NB;
    for (int b = 0; b < NB; ++b) row[NF + b] = bd[b];
    row[103] = 0.0f;                            // K pad (103 -> 104)
    __syncthreads();

    // ---- Phase 2: WMMA over K=104, two M tiles x four N tiles ----
    const int wave = tid >> 5;
    const int lane = tid & 31;
    const int half = lane >> 4;
    const int m    = lane & 15;
    const int r0   = wave * 32;
    const v2f* bbase = wn2 + half * NC + m;

    v8f acc[2][4] = {};
    #pragma unroll                    // straight-line: no acc register rotation
    for (int kk = 0; kk < KE_PAD / 4; ++kk) {
        const int kr0 = kk * 4 + 2 * half;
        v2f a0, a1;
        a0.x = E[(r0 + m)      * RS_E + kr0];
        a0.y = E[(r0 + m)      * RS_E + kr0 + 1];
        a1.x = E[(r0 + 16 + m) * RS_E + kr0];
        a1.y = E[(r0 + 16 + m) * RS_E + kr0 + 1];
        #pragma unroll
        for (int n = 0; n < 4; ++n) {
            const v2f b = bbase[kk * 2 * NC + n * 16];   // one global_load_b64
            acc[0][n] = wmma_f32(a0, b, acc[0][n]);
            acc[1][n] = wmma_f32(a1, b, acc[1][n]);
        }
    }

    // ---- Phase 3: atomic scatter-add onto source nodes ----
    #pragma unroll
    for (int mt = 0; mt < 2; ++mt) {
        #pragma unroll
        for (int v = 0; v < 8; ++v) {
            const int lrow = r0 + mt * 16 + v + 8 * half;
            const int s = Src[lrow];
            #pragma unroll
            for (int n = 0; n < 4; ++n)
                atomic_add_f32(&out[s * NC + n * 16 + m], acc[mt][n][v]);
        }
    }
}

// ---------------------------------------------------------------------------
extern "C" void kernel_launch(void* const* d_in, const int* in_sizes, int n_in,
                              void* d_out, int out_size, void* d_ws, size_t ws_size,
                              hipStream_t stream) {
    const float* feat = (const float*)d_in[0];   // features [N_NODES, NF]
    const float* bond = (const float*)d_in[1];   // bond     [N_EDGES, NB]
    const float* w_s  = (const float*)d_in[2];   // [NF, NC]
    const float* w_n  = (const float*)d_in[3];   // [KE, NC]
    const int*   esrc = (const int*)d_in[4];
    const int*   edst = (const int*)d_in[5];
    float* out = (float*)d_out;

    v2f* ws2 = (v2f*)d_ws;                       // [KS_PAD/2 * NC]
    v2f* wn2 = ws2 + (KS_PAD / 2) * NC;          // [KE_PAD/2 * NC]

    // 0) repack weights into fragment order (one b64 per B fragment)
    pack_weights<<<((KE_PAD / 2) * NC + 255) / 256, 256, 0, stream>>>(w_s, w_n, ws2, wn2);
    // 1) initialize out with self transform
    self_gemm<<<(N_NODES + 63) / 64, 128, 0, stream>>>(feat, ws2, out);
    // 2) accumulate edge messages
    edge_gemm<<<N_EDGES / 128, 128, 0, stream>>>(feat, bond, wn2, esrc, edst, out);
}